// GNNArma_51462298140986
// MI455X (gfx1250) — compile-verified
//
#include <hip/hip_runtime.h>

#define N_NODES 100000
#define N_EDGES 1600000
#define H 32
#define K_ST 3
#define T_LAY 5
#define EPS_GN 1e-5f
#define KNH ((size_t)K_ST * N_NODES * H)

typedef __attribute__((ext_vector_type(2))) float v2f;
typedef __attribute__((ext_vector_type(8))) float v8f;

// ---------------- gcn_norm ----------------
__global__ void deg_kernel(const int* __restrict__ col, const float* __restrict__ w,
                           float* __restrict__ deg) {
    int e = blockIdx.x * blockDim.x + threadIdx.x;
    if (e < N_EDGES) atomicAdd(&deg[col[e]], w[e]);
}

__global__ void dinv_kernel(float* __restrict__ deg) {
    int i = blockIdx.x * blockDim.x + threadIdx.x;
    if (i < N_NODES) {
        float d = deg[i];
        deg[i] = d > 0.f ? rsqrtf(fmaxf(d, 1e-20f)) : 0.f;
    }
}

__global__ void norm_kernel(const int* __restrict__ row, const int* __restrict__ col,
                            const float* __restrict__ w, const float* __restrict__ dinv,
                            float* __restrict__ nrm) {
    int e = blockIdx.x * blockDim.x + threadIdx.x;
    if (e < N_EDGES) nrm[e] = dinv[row[e]] * w[e] * dinv[col[e]];
}

// ---------------- dense transforms ----------------
// Y[k][N,H] = X(k)[N,H] @ W[k][H,H] using V_WMMA_F32_16X16X4_F32 (exact fp32).
// One wave handles a 16-row output tile for one stack. blockDim.x = 128 (4 waves).
__global__ void gemm_nk32_wmma(const float* __restrict__ X, const float* __restrict__ Wm,
                               float* __restrict__ Y, int x_per_stack) {
    const int NT = N_NODES / 16;  // 6250
    int tile = blockIdx.x * (blockDim.x >> 5) + (threadIdx.x >> 5);
    if (tile >= NT) return;  // wave-uniform: EXEC stays all-1s for WMMA
    int k = blockIdx.y;
    const float* Xk = x_per_stack ? (X + (size_t)k * N_NODES * H) : X;
    const float* Wk = Wm + (size_t)k * H * H;
    float* Yk = Y + (size_t)k * N_NODES * H;

    int lane = threadIdx.x & 31;
    int half = lane >> 4;   // K-halves / M-halves selector per ISA layout tables
    int l16  = lane & 15;
    int arow = tile * 16 + l16;

    v8f c0 = {}; v8f c1 = {};
#pragma unroll
    for (int k0 = 0; k0 < H; k0 += 4) {
        // A 16x4 fp32: lanes 0-15 hold K=k0,k0+1 ; lanes 16-31 hold K=k0+2,k0+3
        v2f a = *(const v2f*)(Xk + (size_t)arow * H + k0 + 2 * half);
        int klo = k0 + 2 * half;
        // B 4x16 fp32: per lane {W[klo][n], W[klo+1][n]}, n = column within tile
        v2f b0, b1;
        b0.x = Wk[klo * H + l16];        b0.y = Wk[(klo + 1) * H + l16];
        b1.x = Wk[klo * H + 16 + l16];   b1.y = Wk[(klo + 1) * H + 16 + l16];
        c0 = __builtin_amdgcn_wmma_f32_16x16x4_f32(false, a, false, b0, (short)0, c0, false, false);
        c1 = __builtin_amdgcn_wmma_f32_16x16x4_f32(false, a, false, b1, (short)0, c1, false, false);
    }
#pragma unroll
    for (int j = 0; j < 8; ++j) {
        int r = tile * 16 + j + half * 8;  // C/D: VGPR j <-> row j (lanes 0-15) / j+8
        Yk[(size_t)r * H + l16]      = c0[j];
        Yk[(size_t)r * H + 16 + l16] = c1[j];
    }
}

// F_IN==2 transform (conv1 init/root): Y[k][n][c] = x[n][0]*W[k][0][c] + x[n][1]*W[k][1][c]
__global__ void xform_fin2(const float* __restrict__ x, const float* __restrict__ Wm,
                           float* __restrict__ Y) {
    int i = blockIdx.x * blockDim.x + threadIdx.x;  // over N*H
    if (i >= N_NODES * H) return;
    int k = blockIdx.y;
    int n = i >> 5, c = i & 31;
    Y[(size_t)k * N_NODES * H + i] =
        x[n * 2] * Wm[k * 64 + c] + x[n * 2 + 1] * Wm[k * 64 + 32 + c];
}

// ---------------- propagation ----------------
__global__ void propagate_kernel(const float* __restrict__ src, const int* __restrict__ row,
                                 const int* __restrict__ col, const float* __restrict__ nrm,
                                 float* __restrict__ agg) {
    int i = blockIdx.x * blockDim.x + threadIdx.x;  // E*32 lanes
    int e = i >> 5, c = i & 31;
    if (e >= N_EDGES) return;
    size_t base = (size_t)blockIdx.y * N_NODES * H;
    float v = src[base + (size_t)row[e] * H + c] * nrm[e];
    atomicAdd(&agg[base + (size_t)col[e] * H + c], v);
}

// out = relu(agg + root(in out) + bias[t])
__global__ void combine_relu_kernel(float* __restrict__ out, const float* __restrict__ agg,
                                    const float* __restrict__ bias) {
    int i = blockIdx.x * blockDim.x + threadIdx.x;
    if (i >= N_NODES * H) return;
    int k = blockIdx.y, c = i & 31;
    size_t idx = (size_t)k * N_NODES * H + i;
    float v = agg[idx] + out[idx] + bias[k * H + c];
    out[idx] = v > 0.f ? v : 0.f;
}

// ---------------- GraphNorm ----------------
__global__ void meank_colsum_kernel(const float* __restrict__ out, float* __restrict__ h,
                                    float* __restrict__ sums) {
    __shared__ float s[H];
    int i = blockIdx.x * blockDim.x + threadIdx.x;
    if (threadIdx.x < H) s[threadIdx.x] = 0.f;
    __syncthreads();
    if (i < N_NODES * H) {
        float v = (out[i] + out[(size_t)N_NODES * H + i] + out[2 * (size_t)N_NODES * H + i])
                  * (1.f / 3.f);
        h[i] = v;
        atomicAdd(&s[i & 31], v);
    }
    __syncthreads();
    if (threadIdx.x < H) atomicAdd(&sums[threadIdx.x], s[threadIdx.x]);
}

__global__ void gn_sumsq_kernel(const float* __restrict__ h, const float* __restrict__ sums,
                                const float* __restrict__ gms, float* __restrict__ sumsq) {
    __shared__ float s[H];
    int i = blockIdx.x * blockDim.x + threadIdx.x;
    if (threadIdx.x < H) s[threadIdx.x] = 0.f;
    __syncthreads();
    if (i < N_NODES * H) {
        int c = i & 31;
        float t = h[i] - gms[c] * (sums[c] * (1.f / N_NODES));
        atomicAdd(&s[c], t * t);
    }
    __syncthreads();
    if (threadIdx.x < H) atomicAdd(&sumsq[threadIdx.x], s[threadIdx.x]);
}

__global__ void gn_final_kernel(float* __restrict__ h, const float* __restrict__ stats,
                                const float* __restrict__ gw, const float* __restrict__ gb,
                                const float* __restrict__ gms) {
    int i = blockIdx.x * blockDim.x + threadIdx.x;
    if (i >= N_NODES * H) return;
    int c = i & 31;
    float mean = stats[c] * (1.f / N_NODES);
    float var  = stats[H + c] * (1.f / N_NODES);
    float t = h[i] - gms[c] * mean;
    float y = gw[c] * t * rsqrtf(var + EPS_GN) + gb[c];
    h[i] = y > 0.f ? y : 0.f;
}

// ---------------- classifier ----------------
__global__ void cls_kernel(const float* __restrict__ h, const float* __restrict__ w,
                           const float* __restrict__ b, float* __restrict__ out) {
    int n = blockIdx.x * blockDim.x + threadIdx.x;
    if (n >= N_NODES) return;
    float a0 = b[0], a1 = b[1];
#pragma unroll
    for (int c = 0; c < H; ++c) {
        float v = h[(size_t)n * H + c];
        a0 += v * w[c * 2];
        a1 += v * w[c * 2 + 1];
    }
    out[n * 2] = a0;
    out[n * 2 + 1] = a1;
}

// ---------------- host orchestration ----------------
static void run_conv(const float* xin, bool fin2, const float* initW, const float* Wt,
                     const float* rootW, const float* bias, float* bufO, float* bufT,
                     float* bufA, const int* row, const int* col, const float* nrm,
                     hipStream_t stream) {
    dim3 grdElem(12500, K_ST);              // N*H / 256
    dim3 grdGemm((6250 + 3) / 4, K_ST);     // tiles / waves-per-block
    dim3 grdProp(200000, K_ST);             // E*32 / 256

    // t = 0: bufT = x @ init_weight
    if (fin2) xform_fin2<<<grdElem, 256, 0, stream>>>(xin, initW, bufT);
    else      gemm_nk32_wmma<<<grdGemm, 128, 0, stream>>>(xin, initW, bufT, 0);

    for (int t = 0; t < T_LAY; ++t) {
        if (t > 0)  // bufT = bufO @ weight[t-1]  (per-stack input)
            gemm_nk32_wmma<<<grdGemm, 128, 0, stream>>>(
                bufO, Wt + (size_t)(t - 1) * K_ST * H * H, bufT, 1);
        // root term -> bufO (bufO already consumed by the gemm above)
        if (fin2)
            xform_fin2<<<grdElem, 256, 0, stream>>>(xin, rootW + (size_t)t * K_ST * 2 * H, bufO);
        else
            gemm_nk32_wmma<<<grdGemm, 128, 0, stream>>>(
                xin, rootW + (size_t)t * K_ST * H * H, bufO, 0);
        hipMemsetAsync(bufA, 0, KNH * sizeof(float), stream);
        propagate_kernel<<<grdProp, 256, 0, stream>>>(bufT, row, col, nrm, bufA);
        combine_relu_kernel<<<grdElem, 256, 0, stream>>>(bufO, bufA, bias + (size_t)t * K_ST * H);
    }
}

static void run_gn(const float* bufO, float* h, float* stats, const float* gw, const float* gb,
                   const float* gms, hipStream_t stream) {
    hipMemsetAsync(stats, 0, 2 * H * sizeof(float), stream);
    meank_colsum_kernel<<<12500, 256, 0, stream>>>(bufO, h, stats);
    gn_sumsq_kernel<<<12500, 256, 0, stream>>>(h, stats, gms, stats + H);
    gn_final_kernel<<<12500, 256, 0, stream>>>(h, stats, gw, gb, gms);
}

extern "C" void kernel_launch(void* const* d_in, const int* in_sizes, int n_in,
                              void* d_out, int out_size, void* d_ws, size_t ws_size,
                              hipStream_t stream) {
    const float* x    = (const float*)d_in[0];
    const int*   ei   = (const int*)d_in[1];
    const float* ew   = (const float*)d_in[2];
    const float* c1i  = (const float*)d_in[3];
    const float* c1w  = (const float*)d_in[4];
    const float* c1r  = (const float*)d_in[5];
    const float* c1b  = (const float*)d_in[6];
    const float* c2i  = (const float*)d_in[7];
    const float* c2w  = (const float*)d_in[8];
    const float* c2r  = (const float*)d_in[9];
    const float* c2b  = (const float*)d_in[10];
    const float* c3i  = (const float*)d_in[11];
    const float* c3w  = (const float*)d_in[12];
    const float* c3r  = (const float*)d_in[13];
    const float* c3b  = (const float*)d_in[14];
    const float* g1w  = (const float*)d_in[15];
    const float* g1b  = (const float*)d_in[16];
    const float* g1m  = (const float*)d_in[17];
    const float* g2w  = (const float*)d_in[18];
    const float* g2b  = (const float*)d_in[19];
    const float* g2m  = (const float*)d_in[20];
    const float* g3w  = (const float*)d_in[21];
    const float* g3b  = (const float*)d_in[22];
    const float* g3m  = (const float*)d_in[23];
    const float* clsw = (const float*)d_in[24];
    const float* clsb = (const float*)d_in[25];

    float* ws    = (float*)d_ws;
    float* deg   = ws;                      // N
    float* nrm   = deg + N_NODES;           // E
    float* bufO  = nrm + N_EDGES;           // K*N*H
    float* bufT  = bufO + KNH;              // K*N*H
    float* bufA  = bufT + KNH;              // K*N*H
    float* h     = bufA + KNH;              // N*H
    float* stats = h + (size_t)N_NODES * H; // 2*H

    const int* row = ei;
    const int* col = ei + N_EDGES;

    // gcn_norm
    hipMemsetAsync(deg, 0, N_NODES * sizeof(float), stream);
    deg_kernel<<<(N_EDGES + 255) / 256, 256, 0, stream>>>(col, ew, deg);
    dinv_kernel<<<(N_NODES + 255) / 256, 256, 0, stream>>>(deg);
    norm_kernel<<<(N_EDGES + 255) / 256, 256, 0, stream>>>(row, col, ew, deg, nrm);

    // conv1 (F_IN=2) + gn1
    run_conv(x, true, c1i, c1w, c1r, c1b, bufO, bufT, bufA, row, col, nrm, stream);
    run_gn(bufO, h, stats, g1w, g1b, g1m, stream);
    // conv2 (F_IN=32, WMMA everywhere) + gn2
    run_conv(h, false, c2i, c2w, c2r, c2b, bufO, bufT, bufA, row, col, nrm, stream);
    run_gn(bufO, h, stats, g2w, g2b, g2m, stream);
    // conv3 + gn3
    run_conv(h, false, c3i, c3w, c3r, c3b, bufO, bufT, bufA, row, col, nrm, stream);
    run_gn(bufO, h, stats, g3w, g3b, g3m, stream);

    // classifier
    cls_kernel<<<(N_NODES + 255) / 256, 256, 0, stream>>>(h, clsw, clsb, (float*)d_out);
}